// EEGNetAAD_14199161881122
// MI455X (gfx1250) — compile-verified
//
#include <hip/hip_runtime.h>

typedef __attribute__((ext_vector_type(2))) float v2f;
typedef __attribute__((ext_vector_type(8))) float v8f;

#define NCHN   64
#define EEGL   256
#define BATCH  16
#define LWIN   193          // 256 - 64 + 1
#define F2C    16
#define ZZ_ELEMS (BATCH * F2C * EEGL)   // 65536 floats = 256 KB

// ---------------------------------------------------------------------------
// Kernel 0: fold BN / bias constants into a packed param block P (513 floats)
// P: [0]T0[16] [16]T1 [32]T2 [48]Cfold [64]w3k0 [80]w3k1 [96]w3k2 [112]b3
//    [128]w4[256] [384]s3[16] [400]c3[16] [416]lin_w[96] [512]lin_b
// ---------------------------------------------------------------------------
__global__ void eegnet_setup(const float* __restrict__ conv1_w, const float* __restrict__ conv1_b,
                             const float* __restrict__ bn1_g, const float* __restrict__ bn1_b,
                             const float* __restrict__ bn1_m, const float* __restrict__ bn1_v,
                             const float* __restrict__ conv2_w, const float* __restrict__ conv2_b,
                             const float* __restrict__ bn2_g, const float* __restrict__ bn2_b,
                             const float* __restrict__ bn2_m, const float* __restrict__ bn2_v,
                             const float* __restrict__ conv3_w, const float* __restrict__ conv3_b,
                             const float* __restrict__ conv4_w, const float* __restrict__ conv4_b,
                             const float* __restrict__ bn3_g, const float* __restrict__ bn3_b,
                             const float* __restrict__ bn3_m, const float* __restrict__ bn3_v,
                             const float* __restrict__ lin_w, const float* __restrict__ lin_b,
                             float* __restrict__ P)
{
    int t = threadIdx.x;
    if (t < 16) {
        int f1 = t >> 1;
        float s1 = bn1_g[f1] * rsqrtf(bn1_v[f1] + 1e-5f);
        float c1 = s1 * (conv1_b[f1] - bn1_m[f1]) + bn1_b[f1];
        float s2 = bn2_g[t] * rsqrtf(bn2_v[t] + 1e-5f);
        float w2sum = 0.f;
        for (int h = 0; h < NCHN; ++h) w2sum += conv2_w[t * NCHN + h];
        float ss = s2 * s1;
        P[0  + t] = ss * conv1_w[f1 * 3 + 0];
        P[16 + t] = ss * conv1_w[f1 * 3 + 1];
        P[32 + t] = ss * conv1_w[f1 * 3 + 2];
        P[48 + t] = s2 * (c1 * w2sum + conv2_b[t] - bn2_m[t]) + bn2_b[t];
        P[64  + t] = conv3_w[t * 3 + 0];
        P[80  + t] = conv3_w[t * 3 + 1];
        P[96  + t] = conv3_w[t * 3 + 2];
        P[112 + t] = conv3_b[t];
        float s3 = bn3_g[t] * rsqrtf(bn3_v[t] + 1e-5f);
        P[384 + t] = s3;
        P[400 + t] = s3 * (conv4_b[t] - bn3_m[t]) + bn3_b[t];
    }
    for (int i = t; i < 256; i += blockDim.x) P[128 + i] = conv4_w[i];
    for (int i = t; i < 96;  i += blockDim.x) P[416 + i] = lin_w[i];
    if (t == 0) P[512] = lin_b[0];
}

// ---------------------------------------------------------------------------
// Kernel 1: ZZ[b][f][u] = sum_h W2[f][h] * eeg[b][h][u]  via V_WMMA_F32_16X16X4_F32
// One wave per (batch b, 16-wide time tile u0): M=16 filters, K=64 chans, N=16 times.
// ---------------------------------------------------------------------------
__global__ void eegnet_zz_wmma(const float* __restrict__ eeg,
                               const float* __restrict__ conv2_w,
                               float* __restrict__ ZZ)
{
    int wid  = (blockIdx.x * blockDim.x + threadIdx.x) >> 5;  // 0..255
    int lane = threadIdx.x & 31;
    int b  = wid >> 4;
    int u0 = (wid & 15) << 4;
    int m    = lane & 15;            // M row (filter) / N col (time)
    int koff = (lane >> 4) << 1;     // lanes 16-31 carry K+2 per 32-bit A/B layout

    const float* eegb = eeg + (size_t)b * (NCHN * EEGL) + u0 + m;
    __builtin_prefetch(eegb, 0, 1);  // global_prefetch_b8

    // A = W2 (16x64): lane m holds row m; k-step s covers K = 4s..4s+3
    v2f a[16];
#pragma unroll
    for (int s = 0; s < 16; ++s) {
        int k = 4 * s + koff;
        a[s].x = conv2_w[m * NCHN + k];
        a[s].y = conv2_w[m * NCHN + k + 1];
    }

    v8f acc = {};
#pragma unroll
    for (int s = 0; s < 16; ++s) {
        int k = 4 * s + koff;
        v2f bv;
        bv.x = eegb[(size_t)k * EEGL];
        bv.y = eegb[(size_t)(k + 1) * EEGL];
        acc = __builtin_amdgcn_wmma_f32_16x16x4_f32(
            /*neg_a=*/false, a[s], /*neg_b=*/false, bv,
            /*c_mod=*/(short)0, acc, /*reuse_a=*/false, /*reuse_b=*/false);
    }

    // D 16x16 f32 layout: VGPR r -> M = r (lanes 0-15) / r+8 (lanes 16-31), N = lane%16
    int fbase = (lane >> 4) << 3;
#pragma unroll
    for (int r = 0; r < 8; ++r) {
        int f = fbase + r;
        ZZ[(size_t)b * (F2C * EEGL) + f * EEGL + u0 + m] = acc[r];
    }
}

// ---------------------------------------------------------------------------
// Kernel 2: one wave per window (b,l). Lane t owns times 2t,2t+1 -> pool2 local.
// conv3 via shfl neighbors, conv4 lane-local 16x16, pool5+linear via shfl-xor.
// ---------------------------------------------------------------------------
__global__ void eegnet_windows(const float* __restrict__ ZZ,
                               const float* __restrict__ P,
                               float* __restrict__ out)
{
    int wid = blockIdx.x * (blockDim.x >> 5) + (threadIdx.x >> 5);
    if (wid >= BATCH * LWIN) return;
    int lane = threadIdx.x & 31;
    int b = wid / LWIN;
    int l = wid - b * LWIN;

    const float* zb = ZZ + (size_t)b * (F2C * EEGL) + l;
    int t2 = lane << 1;                       // window-relative even time

    // conv1(taps over Z) + folded bn1/bn2 + ELU + avgpool(2)
    float pf[16];
#pragma unroll
    for (int f = 0; f < 16; ++f) {
        const float* z = zb + f * EEGL;
        float v0 = (t2 > 0)      ? z[t2 - 1] : 0.f;   // conv1 zero-pad at t=-1
        float v1 = z[t2];
        float v2 = z[t2 + 1];
        float v3 = (t2 + 2 < 64) ? z[t2 + 2] : 0.f;   // conv1 zero-pad at t=64
        float T0 = P[f], T1 = P[16 + f], T2 = P[32 + f], Cf = P[48 + f];
        float ya = fmaf(T0, v0, fmaf(T1, v1, fmaf(T2, v2, Cf)));
        float yb = fmaf(T0, v1, fmaf(T1, v2, fmaf(T2, v3, Cf)));
        ya = ya > 0.f ? ya : __expf(ya) - 1.f;
        yb = yb > 0.f ? yb : __expf(yb) - 1.f;
        pf[f] = 0.5f * (ya + yb);
    }

    // conv3: depthwise 1x3, same padding over 32 positions (lane = position)
    float q[16];
#pragma unroll
    for (int f = 0; f < 16; ++f) {
        float left  = __shfl(pf[f], lane - 1, 32);
        float right = __shfl(pf[f], lane + 1, 32);
        if (lane == 0)  left  = 0.f;
        if (lane == 31) right = 0.f;
        q[f] = fmaf(P[64 + f], left,
               fmaf(P[80 + f], pf[f],
               fmaf(P[96 + f], right, P[112 + f])));
    }

    // conv4 (16x16 pointwise, lane-local) + bn3 + ELU + pool5/linear partials
    float acc = 0.f;
    int i6 = lane / 5; if (i6 > 5) i6 = 5;
#pragma unroll
    for (int g = 0; g < 16; ++g) {
        float r = 0.f;
#pragma unroll
        for (int f = 0; f < 16; ++f) r = fmaf(P[128 + g * 16 + f], q[f], r);
        float e = fmaf(P[384 + g], r, P[400 + g]);
        e = e > 0.f ? e : __expf(e) - 1.f;
        if (lane < 30) acc = fmaf(e, P[416 + g * 6 + i6], acc);
    }

#pragma unroll
    for (int off = 16; off > 0; off >>= 1) acc += __shfl_xor(acc, off, 32);
    if (lane == 0) out[b * LWIN + l] = fmaf(0.2f, acc, P[512]);
}

// ---------------------------------------------------------------------------
extern "C" void kernel_launch(void* const* d_in, const int* in_sizes, int n_in,
                              void* d_out, int out_size, void* d_ws, size_t ws_size,
                              hipStream_t stream) {
    (void)in_sizes; (void)n_in; (void)out_size; (void)ws_size;
    const float* eeg     = (const float*)d_in[0];
    const float* conv1_w = (const float*)d_in[2];
    const float* conv1_b = (const float*)d_in[3];
    const float* bn1_g   = (const float*)d_in[4];
    const float* bn1_b   = (const float*)d_in[5];
    const float* bn1_m   = (const float*)d_in[6];
    const float* bn1_v   = (const float*)d_in[7];
    const float* conv2_w = (const float*)d_in[8];
    const float* conv2_b = (const float*)d_in[9];
    const float* bn2_g   = (const float*)d_in[10];
    const float* bn2_b   = (const float*)d_in[11];
    const float* bn2_m   = (const float*)d_in[12];
    const float* bn2_v   = (const float*)d_in[13];
    const float* conv3_w = (const float*)d_in[14];
    const float* conv3_b = (const float*)d_in[15];
    const float* conv4_w = (const float*)d_in[16];
    const float* conv4_b = (const float*)d_in[17];
    const float* bn3_g   = (const float*)d_in[18];
    const float* bn3_b   = (const float*)d_in[19];
    const float* bn3_m   = (const float*)d_in[20];
    const float* bn3_v   = (const float*)d_in[21];
    const float* lin_w   = (const float*)d_in[22];
    const float* lin_b   = (const float*)d_in[23];

    float* ZZ = (float*)d_ws;             // 65536 floats
    float* P  = ZZ + ZZ_ELEMS;            // 513 floats

    eegnet_setup<<<1, 256, 0, stream>>>(conv1_w, conv1_b, bn1_g, bn1_b, bn1_m, bn1_v,
                                        conv2_w, conv2_b, bn2_g, bn2_b, bn2_m, bn2_v,
                                        conv3_w, conv3_b, conv4_w, conv4_b,
                                        bn3_g, bn3_b, bn3_m, bn3_v, lin_w, lin_b, P);
    eegnet_zz_wmma<<<32, 256, 0, stream>>>(eeg, conv2_w, ZZ);
    eegnet_windows<<<386, 256, 0, stream>>>(ZZ, P, (float*)d_out);
}